// ChebEdgeClassifier_40037685133540
// MI455X (gfx1250) — compile-verified
//
#include <hip/hip_runtime.h>
#include <hip/hip_bf16.h>
#include <stdint.h>

typedef float v2f __attribute__((ext_vector_type(2)));
typedef float v8f __attribute__((ext_vector_type(8)));

#define CH 128  // IN == H == 128

// ---------------------------------------------------------------------------
// small elementwise kernels
// ---------------------------------------------------------------------------
__global__ void k_zero(float* __restrict__ p, long long n) {
  long long i = (long long)blockIdx.x * blockDim.x + threadIdx.x;
  if (i < n) p[i] = 0.0f;
}

__global__ void k_neg_copy(const float* __restrict__ s, float* __restrict__ d, long long n) {
  long long i = (long long)blockIdx.x * blockDim.x + threadIdx.x;
  if (i < n) d[i] = -s[i];
}

__global__ void k_deg(const int* __restrict__ src, const float* __restrict__ w,
                      float* __restrict__ deg, int E) {
  int e = blockIdx.x * blockDim.x + threadIdx.x;
  if (e < E) atomicAdd(&deg[src[e]], w[e]);
}

__global__ void k_dinv(const float* __restrict__ deg, float* __restrict__ dinv, int N) {
  int i = blockIdx.x * blockDim.x + threadIdx.x;
  if (i < N) {
    float d = deg[i];
    dinv[i] = (d > 0.0f) ? rsqrtf(d) : 0.0f;
  }
}

__global__ void k_norm(const int* __restrict__ src, const int* __restrict__ dst,
                       const float* __restrict__ w, const float* __restrict__ dinv,
                       float* __restrict__ nrm, int E) {
  int e = blockIdx.x * blockDim.x + threadIdx.x;
  if (e < E) nrm[e] = -dinv[src[e]] * w[e] * dinv[dst[e]];
}

// ---------------------------------------------------------------------------
// SpMM scatter: out[dst[e], :] += scale * norm[e] * h[src[e], :]
// One wave per edge: 32 lanes * float4 == full 128-float row (512B coalesced).
// norm/src/dst are wave-uniform -> scalar loads.
// ---------------------------------------------------------------------------
__global__ void k_prop(const float* __restrict__ h, const float* __restrict__ nrm,
                       const int* __restrict__ src, const int* __restrict__ dst,
                       float* __restrict__ out, float scale, int E) {
  long long idx = (long long)blockIdx.x * blockDim.x + threadIdx.x;
  int e = (int)(idx >> 5);
  if (e >= E) return;
  int lane = (int)(idx & 31);
  float coef = scale * nrm[e];
  const float4 hv = *(const float4*)(h + (long long)src[e] * CH + lane * 4);
  float* o = out + (long long)dst[e] * CH + lane * 4;
  atomicAdd(o + 0, coef * hv.x);
  atomicAdd(o + 1, coef * hv.y);
  atomicAdd(o + 2, coef * hv.z);
  atomicAdd(o + 3, coef * hv.w);
}

// ---------------------------------------------------------------------------
// Fused Chebyshev GEMM:  out = t0@W[0] + t1@W[1] + t2@W[2] + bias  (opt. ReLU)
// fp32 WMMA path: v_wmma_f32_16x16x4_f32.
// Block = 256 threads (8 waves); wave w computes rows [blk*128 + 16w, +16).
// Weight matrix staged in LDS as float2 {W[2k][n], W[2k+1][n]} so each lane's
// B fragment is a single 8-byte LDS load matching the ISA B layout.
// Tail handling: A-row index is CLAMPED (not predicated) so the hot loop is
// branch-free with EXEC all-ones; out-of-range rows compute garbage that the
// store guard in the epilogue discards.
// ---------------------------------------------------------------------------
__global__ __launch_bounds__(256) void k_cheb_gemm(
    const float* __restrict__ t0, const float* __restrict__ t1,
    const float* __restrict__ t2, const float* __restrict__ W,
    const float* __restrict__ bias, float* __restrict__ out,
    int nrows, int relu) {
  __shared__ float2 WL[64 * 128];  // 64 KB: k-pair-interleaved 128x128 weights

  const int tid = threadIdx.x;
  const int wave = tid >> 5;
  const int lane = tid & 31;
  const int m = lane & 15;     // row-within-tile for A, col-within-tile for B/C
  const int kh = lane >> 4;    // lane-half selects K pair (A/B) and M half (C)
  const int rowBase = blockIdx.x * 128 + wave * 16;

  // clamp instead of predicate: keeps inner loop branch-free
  int rowA = rowBase + m;
  if (rowA > nrows - 1) rowA = nrows - 1;

  v8f acc[8] = {};  // 8 col-tiles x 16x16 f32 accumulators

  const float* tx[3] = {t0, t1, t2};

  for (int kk = 0; kk < 3; ++kk) {
    __syncthreads();
    // cooperative stage of W[kk] into LDS, interleaving k-pairs
    const float* Wk = W + kk * CH * CH;
    for (int i = tid; i < 64 * 128; i += 256) {
      int k2 = i >> 7;
      int n = i & 127;
      WL[i] = make_float2(Wk[(2 * k2) * CH + n], Wk[(2 * k2 + 1) * CH + n]);
    }
    __syncthreads();

    const float* Arow = tx[kk] + (long long)rowA * CH + 2 * kh;

    for (int ks = 0; ks < 32; ++ks) {  // K = 128 in steps of 4
      float2 av = *(const float2*)(Arow + 4 * ks);
      v2f a;
      a.x = av.x; a.y = av.y;
      const float2* Brow = &WL[(2 * ks + kh) * 128];
#pragma unroll
      for (int t = 0; t < 8; ++t) {
        float2 bv = Brow[t * 16 + m];
        v2f b; b.x = bv.x; b.y = bv.y;
        acc[t] = __builtin_amdgcn_wmma_f32_16x16x4_f32(
            false, a, false, b, (short)0, acc[t], false, false);
      }
    }
  }

  // epilogue: C tile layout -> row = rowBase + v + 8*kh, col = 16t + m
#pragma unroll
  for (int t = 0; t < 8; ++t) {
    const int col = t * 16 + m;
    const float bb = bias[col];
#pragma unroll
    for (int v = 0; v < 8; ++v) {
      const int row = rowBase + v + 8 * kh;
      if (row < nrows) {
        float val = acc[t][v] + bb;
        if (relu) val = fmaxf(val, 0.0f);
        out[(long long)row * CH + col] = val;
      }
    }
  }
}

// ---------------------------------------------------------------------------
// Edge classifier: out[e] = concat(h[src], h[dst]) @ Wc + bc   (Wc: [256,2])
// ---------------------------------------------------------------------------
__global__ void k_edge_out(const float* __restrict__ h, const int* __restrict__ src,
                           const int* __restrict__ dst, const float* __restrict__ Wc,
                           const float* __restrict__ bc, float* __restrict__ out, int E) {
  int e = blockIdx.x * blockDim.x + threadIdx.x;
  if (e >= E) return;
  const float* hs = h + (long long)src[e] * CH;
  const float* hd = h + (long long)dst[e] * CH;
  float o0 = bc[0], o1 = bc[1];
#pragma unroll 4
  for (int c = 0; c < CH; c += 4) {
    float4 a = *(const float4*)(hs + c);
    float4 b = *(const float4*)(hd + c);
    o0 += a.x * Wc[(c + 0) * 2 + 0] + a.y * Wc[(c + 1) * 2 + 0] +
          a.z * Wc[(c + 2) * 2 + 0] + a.w * Wc[(c + 3) * 2 + 0];
    o1 += a.x * Wc[(c + 0) * 2 + 1] + a.y * Wc[(c + 1) * 2 + 1] +
          a.z * Wc[(c + 2) * 2 + 1] + a.w * Wc[(c + 3) * 2 + 1];
    o0 += b.x * Wc[(CH + c + 0) * 2 + 0] + b.y * Wc[(CH + c + 1) * 2 + 0] +
          b.z * Wc[(CH + c + 2) * 2 + 0] + b.w * Wc[(CH + c + 3) * 2 + 0];
    o1 += b.x * Wc[(CH + c + 0) * 2 + 1] + b.y * Wc[(CH + c + 1) * 2 + 1] +
          b.z * Wc[(CH + c + 2) * 2 + 1] + b.w * Wc[(CH + c + 3) * 2 + 1];
  }
  out[(long long)e * 2 + 0] = o0;
  out[(long long)e * 2 + 1] = o1;
}

// ---------------------------------------------------------------------------
extern "C" void kernel_launch(void* const* d_in, const int* in_sizes, int n_in,
                              void* d_out, int out_size, void* d_ws, size_t ws_size,
                              hipStream_t stream) {
  const float* x  = (const float*)d_in[0];
  const int*   ei = (const int*)d_in[1];
  const float* w  = (const float*)d_in[2];
  const float* W1 = (const float*)d_in[3];
  const float* b1 = (const float*)d_in[4];
  const float* W2 = (const float*)d_in[5];
  const float* b2 = (const float*)d_in[6];
  const float* Wc = (const float*)d_in[7];
  const float* bc = (const float*)d_in[8];
  float* out = (float*)d_out;

  const int N = in_sizes[0] / CH;
  const int E = in_sizes[2];
  const int* src = ei;
  const int* dst = ei + E;

  // workspace carve-up (256B aligned)
  char* ws = (char*)d_ws;
  size_t off = 0;
  auto carve = [&](size_t bytes) -> float* {
    float* p = (float*)(ws + off);
    off = (off + bytes + 255) & ~(size_t)255;
    return p;
  };
  float* deg  = carve((size_t)N * 4);
  float* dinv = carve((size_t)N * 4);
  float* nrm  = carve((size_t)E * 4);
  float* bufA = carve((size_t)N * CH * 4);  // Tx1 / u1
  float* bufB = carve((size_t)N * CH * 4);  // Tx2 / u2
  float* h1   = carve((size_t)N * CH * 4);
  float* h2   = carve((size_t)N * CH * 4);
  (void)ws_size; (void)n_in; (void)out_size;

  const int B = 256;
  const long long NC = (long long)N * CH;
  const int gNC = (int)((NC + B - 1) / B);
  const int gN  = (N + B - 1) / B;
  const int gE  = (E + B - 1) / B;
  const long long propWork = (long long)E * 32;
  const int gProp = (int)((propWork + B - 1) / B);
  const int gGemm = (N + 127) / 128;

  // graph normalization
  k_zero<<<gN, B, 0, stream>>>(deg, N);
  k_deg<<<gE, B, 0, stream>>>(src, w, deg, E);
  k_dinv<<<gN, B, 0, stream>>>(deg, dinv, N);
  k_norm<<<gE, B, 0, stream>>>(src, dst, w, dinv, nrm, E);

  // ---- layer 1: h1 = relu(x@W1_0 + T1@W1_1 + T2@W1_2 + b1)
  k_zero<<<gNC, B, 0, stream>>>(bufA, NC);
  k_prop<<<gProp, B, 0, stream>>>(x, nrm, src, dst, bufA, 1.0f, E);      // T1 = L~ x
  k_neg_copy<<<gNC, B, 0, stream>>>(x, bufB, NC);                        // T2 := -x
  k_prop<<<gProp, B, 0, stream>>>(bufA, nrm, src, dst, bufB, 2.0f, E);   // T2 += 2 L~ T1
  k_cheb_gemm<<<gGemm, 256, 0, stream>>>(x, bufA, bufB, W1, b1, h1, N, 1);

  // ---- layer 2: h2 = h1@W2_0 + U1@W2_1 + U2@W2_2 + b2
  k_zero<<<gNC, B, 0, stream>>>(bufA, NC);
  k_prop<<<gProp, B, 0, stream>>>(h1, nrm, src, dst, bufA, 1.0f, E);
  k_neg_copy<<<gNC, B, 0, stream>>>(h1, bufB, NC);
  k_prop<<<gProp, B, 0, stream>>>(bufA, nrm, src, dst, bufB, 2.0f, E);
  k_cheb_gemm<<<gGemm, 256, 0, stream>>>(h1, bufA, bufB, W2, b2, h2, N, 0);

  // ---- edge classifier
  k_edge_out<<<gE, B, 0, stream>>>(h2, src, dst, Wc, bc, out, E);
}